// EdgeLabelPredictor_9852654977718
// MI455X (gfx1250) — compile-verified
//
#include <hip/hip_runtime.h>

typedef __attribute__((ext_vector_type(2))) float v2f;
typedef __attribute__((ext_vector_type(4))) float v4f;
typedef __attribute__((ext_vector_type(8))) float v8f;

#define DF 128          // feature dim
#define DD 4            // dot-product dim
#define WSTRIDE 132     // LDS row stride: 132 mod 64 == 4 -> each of the 16 rows
                        // owns a private 4-bank slot => conflict-free ds_load_b64

// ---------------------------------------------------------------------------
// Projection: out[r][j] = sum_k X[r][k] * W[j][k]   (X: rows x 128, W: 4 x 128)
// One wave32 per 16-row tile, K looped in chunks of 4 via V_WMMA_F32_16X16X4_F32.
//
// A (16x4 f32) VGPR layout (ISA 7.12.2): lane l<16 -> M=l, holds K={0,1};
// lanes 16-31 -> M=l-16, hold K={2,3}.  B (4x16) mirrored with N along lanes.
// D (16x16 f32): VGPR r -> M=r (lanes 0-15, N=lane) / M=8+r (lanes 16-31).
//
// B fragments come from a zero-padded 16x128 copy of W staged in LDS once per
// block: unconditional ds_load_b64 in the loop, no predicated global loads.
// ---------------------------------------------------------------------------
__global__ __launch_bounds__(256) void proj_wmma_f32(const float* __restrict__ X,
                                                     const float* __restrict__ W,
                                                     float* __restrict__ out,
                                                     int rows, int ntiles)
{
    __shared__ float sW[16 * WSTRIDE];

    // Cooperative fill: rows 0..3 = W, rows 4..15 (and pad cols) = 0.
    for (int t = threadIdx.x; t < 16 * WSTRIDE; t += blockDim.x) {
        const int row = t / WSTRIDE;
        const int col = t - row * WSTRIDE;
        sW[t] = (row < DD && col < DF) ? W[row * DF + col] : 0.f;
    }
    __syncthreads();

    const int wave = (int)((blockIdx.x * blockDim.x + threadIdx.x) >> 5); // wave-uniform
    const int lane = (int)(threadIdx.x & 31);
    if (wave >= ntiles) return;   // whole wave exits together -> EXEC all-1s when active

    const int mrow = lane & 15;   // M for A-loads, N (=W row) for B-loads
    const int hi   = lane >> 4;   // lane-half selects K pair {0,1} vs {2,3}

    int grow = wave * 16 + mrow;
    if (grow >= rows) grow = rows - 1;          // clamp keeps all lanes active
    const float* __restrict__ xrow  = X + (size_t)grow * DF;
    const float* __restrict__ wbase = &sW[mrow * WSTRIDE + 2 * hi];

    v8f acc = {0.f, 0.f, 0.f, 0.f, 0.f, 0.f, 0.f, 0.f};

    #pragma unroll
    for (int kk = 0; kk < DF / 4; ++kk) {
        const int k0 = kk * 4 + 2 * hi;
        v2f a;
        a.x = xrow[k0];
        a.y = xrow[k0 + 1];
        const v2f bm = *(const v2f*)(wbase + kk * 4);   // ds_load_b64, imm offset
        // 8 args: (neg_a, A, neg_b, B, c_mod, C, reuse_a, reuse_b)
        acc = __builtin_amdgcn_wmma_f32_16x16x4_f32(false, a, false, bm,
                                                    (short)0, acc, false, false);
    }

    // Write the 4 useful columns: lane with N<4 owns 8 rows (one per D VGPR).
    if (mrow < DD) {
        const int baseRow = wave * 16 + hi * 8;
        #pragma unroll
        for (int r = 0; r < 8; ++r) {
            const int rr = baseRow + r;
            if (rr < rows) out[(size_t)rr * DD + mrow] = acc[r];
        }
    }
}

// ---------------------------------------------------------------------------
// Edge phase: hs = 0.5*(hn[src]+he) + b ; hd = 0.5*(hn[dst]+he) + b ;
// out = sigmoid(hs . hd).  16B gathers per endpoint, streaming on he.
// ---------------------------------------------------------------------------
__global__ __launch_bounds__(256) void edge_score(const float* __restrict__ hn,
                                                  const float* __restrict__ he,
                                                  const int*   __restrict__ src,
                                                  const int*   __restrict__ dst,
                                                  const float* __restrict__ bias,
                                                  float* __restrict__ out,
                                                  int nedges)
{
    const int i = (int)(blockIdx.x * blockDim.x + threadIdx.x);
    if (i >= nedges) return;

    const v4f* __restrict__ hn4 = (const v4f*)hn;
    const v4f* __restrict__ he4 = (const v4f*)he;

    const v4f s = hn4[src[i]];
    const v4f d = hn4[dst[i]];
    const v4f e = he4[i];
    const v4f b = *(const v4f*)bias;

    float dot = 0.f;
    #pragma unroll
    for (int j = 0; j < 4; ++j) {
        const float hs = 0.5f * (s[j] + e[j]) + b[j];
        const float hd = 0.5f * (d[j] + e[j]) + b[j];
        dot += hs * hd;
    }
    out[i] = 1.0f / (1.0f + __expf(-dot));
}

extern "C" void kernel_launch(void* const* d_in, const int* in_sizes, int n_in,
                              void* d_out, int out_size, void* d_ws, size_t ws_size,
                              hipStream_t stream)
{
    const float* n    = (const float*)d_in[0];   // [N_NODES, 128]
    const float* e    = (const float*)d_in[1];   // [N_EDGES, 128]
    const float* W    = (const float*)d_in[2];   // [4, 128]
    const float* bias = (const float*)d_in[3];   // [4]
    const int*   src  = (const int*)d_in[4];     // [N_EDGES]
    const int*   dst  = (const int*)d_in[5];     // [N_EDGES]
    float*       out  = (float*)d_out;           // [N_EDGES, 1]

    const int nnodes = in_sizes[0] / DF;
    const int nedges = in_sizes[1] / DF;

    // Workspace: hn [nnodes,4] then he [nedges,4], 256B-aligned split.
    float* hn = (float*)d_ws;
    size_t hn_elems = ((size_t)nnodes * DD + 63) & ~(size_t)63;
    float* he = hn + hn_elems;

    const int tiles_n = (nnodes + 15) / 16;
    const int tiles_e = (nedges + 15) / 16;
    const dim3 blk(256);   // 8 wave32 per block

    proj_wmma_f32<<<dim3((tiles_n + 7) / 8), blk, 0, stream>>>(n, W, hn, nnodes, tiles_n);
    proj_wmma_f32<<<dim3((tiles_e + 7) / 8), blk, 0, stream>>>(e, W, he, nedges, tiles_e);
    edge_score<<<dim3((nedges + 255) / 256), blk, 0, stream>>>(hn, he, src, dst, bias, out, nedges);
}